// bayesian_categorical_crossentropy_28235115004319
// MI455X (gfx1250) — compile-verified
//
#include <hip/hip_runtime.h>
#include <stdint.h>

#define EPS_F 1e-6f
#define ROWS 8          // waves per block (wave32)
#define CMAX 1000       // reference C

typedef __attribute__((ext_vector_type(4))) unsigned int v4u;
typedef __attribute__((ext_vector_type(8))) int          v8i;
typedef __attribute__((ext_vector_type(4))) int          v4i;

// Low 32 bits of a generic pointer to __shared__ == LDS byte offset
// (flat LDS aperture: addr[31:0] is the LDS address, ISA 10.2).
__device__ __forceinline__ unsigned lds_offset_of(const void* p) {
  return (unsigned)(unsigned long long)(uintptr_t)p;
}

// Issue one TDM row load: nelem f32 elements, global -> LDS.
// D# packing per CDNA5 ISA:
//   group0: count=1, lds_addr, global_addr[56:0], type=2
//   group1: data_size=2 (4B), tensor_dim0=nelem, tensor_dim1=1,
//           tile_dim0=nelem, tile_dim1=1, dim0_stride=nelem, dim1_stride=nelem
//   remaining descriptor words: zero (<= 2-D tensor)
__device__ __forceinline__ void tdm_load_row(unsigned lds_addr,
                                             unsigned long long gaddr,
                                             int nelem) {
  unsigned la  = (unsigned)__builtin_amdgcn_readfirstlane((int)lds_addr);
  unsigned glo = (unsigned)__builtin_amdgcn_readfirstlane((int)(unsigned)(gaddr & 0xFFFFFFFFull));
  unsigned ghi = (unsigned)__builtin_amdgcn_readfirstlane((int)(unsigned)(gaddr >> 32));
  unsigned ne  = (unsigned)__builtin_amdgcn_readfirstlane(nelem);

  v4u g0;
  g0[0] = 1u;                                   // count=1, user mode
  g0[1] = la;                                   // lds_addr (bytes)
  g0[2] = glo;                                  // global_addr[31:0]
  g0[3] = (ghi & 0x1FFFFFFu) | (2u << 30);      // global_addr[56:32] | type=2

  v8i g1;
  g1[0] = (int)(2u << 16);                      // data_size = 4 bytes
  g1[1] = (int)((ne & 0xFFFFu) << 16);          // tensor_dim0[15:0]
  g1[2] = (int)((ne >> 16) | (1u << 16));       // tensor_dim0[31:16] | tensor_dim1=1
  g1[3] = (int)((ne & 0xFFFFu) << 16);          // tile_dim0 = nelem
  g1[4] = 1;                                    // tile_dim1=1, tile_dim2=0
  g1[5] = (int)ne;                              // tensor_dim0_stride[31:0]
  g1[6] = (int)((ne & 0xFFFFu) << 16);          // stride0[47:32]=0 | dim1_stride[15:0]
  g1[7] = (int)(ne >> 16);                      // dim1_stride[47:16]

  v4i z4a = {0, 0, 0, 0};                       // group 2 (unused, 2-D)
  v4i z4b = {0, 0, 0, 0};                       // group 3 (unused, 2-D)
  v8i z8  = {0, 0, 0, 0, 0, 0, 0, 0};           // extra D# words (unused)
  __builtin_amdgcn_tensor_load_to_lds(g0, g1, z4a, z4b, z8, 0);
}

__device__ __forceinline__ float wave_max32(float v) {
  #pragma unroll
  for (int m = 16; m >= 1; m >>= 1) v = fmaxf(v, __shfl_xor(v, m, 32));
  return v;
}
__device__ __forceinline__ float wave_sum32(float v) {
  #pragma unroll
  for (int m = 16; m >= 1; m >>= 1) v += __shfl_xor(v, m, 32);
  return v;
}

// ---------------------------------------------------------------------------
// d_ws layout (floats): [0]=sum_b(lse_u - tgt_u)  [1]=sum_b(exp(var)-1)
//                       [2..2+T) = sum_b(lse_t - tgt_t) per t
// ---------------------------------------------------------------------------

__global__ void k_init(float* __restrict__ ws, int n) {
  int i = blockIdx.x * blockDim.x + threadIdx.x;
  if (i < n) ws[i] = 0.0f;
}

// One wave per (t,b) row. blockIdx.x = bchunk*T + t  (pred L2 reuse across t).
__global__ __launch_bounds__(256) void k_distorted(
    const float* __restrict__ pred_var, const float* __restrict__ noise,
    const int* __restrict__ tru, float* __restrict__ ws,
    int B, int C, int T) {
  __shared__ __align__(16) float s_noise[ROWS * CMAX];
  __shared__ __align__(16) float s_pred [ROWS * CMAX];
  __shared__ float s_red[ROWS];

  const int tid  = threadIdx.x;
  const int lane = tid & 31;
  const int wave = tid >> 5;
  const int t      = blockIdx.x % T;
  const int bchunk = blockIdx.x / T;
  const int b  = bchunk * ROWS + wave;
  const int C1 = C + 1;

  float* myn = &s_noise[wave * C];
  float* myp = &s_pred [wave * C];

  // Async DMA of this wave's noise row and pred row into LDS (TENSORcnt).
  unsigned long long noff =
      ((unsigned long long)((size_t)t * (size_t)B + (size_t)b)) * (unsigned)C * 4ull;
  tdm_load_row(lds_offset_of(myn), (unsigned long long)(uintptr_t)noise + noff, C);
  unsigned long long poff = (unsigned long long)b * (unsigned)C1 * 4ull;
  tdm_load_row(lds_offset_of(myp), (unsigned long long)(uintptr_t)pred_var + poff, C);

  // Overlap: fetch per-row variance while the DMA runs.
  const float var = pred_var[(size_t)b * C1 + C];
  const float sd  = sqrtf(var) + EPS_F;

  __builtin_amdgcn_s_wait_tensorcnt(0);
  asm volatile("" ::: "memory");

  const float4* vn = (const float4*)myn;      // rows are 16B-aligned (wave*4*C)
  const float4* vp = (const float4*)myp;
  const int nv4 = C >> 2;

  // Pass 1: row max of distorted logits (ds_load_b128 x2 per 4 elements).
  float m = -INFINITY;
  for (int i = lane; i < nv4; i += 32) {
    float4 a = vn[i], p = vp[i];
    m = fmaxf(m, fmaf(a.x, sd, p.x));
    m = fmaxf(m, fmaf(a.y, sd, p.y));
    m = fmaxf(m, fmaf(a.z, sd, p.z));
    m = fmaxf(m, fmaf(a.w, sd, p.w));
  }
  for (int i = (nv4 << 2) + lane; i < C; i += 32)   // tail (C%4 != 0)
    m = fmaxf(m, fmaf(myn[i], sd, myp[i]));
  m = wave_max32(m);

  // Pass 2: sum of exp(logit - max).
  float s = 0.0f;
  for (int i = lane; i < nv4; i += 32) {
    float4 a = vn[i], p = vp[i];
    s += __expf(fmaf(a.x, sd, p.x) - m);
    s += __expf(fmaf(a.y, sd, p.y) - m);
    s += __expf(fmaf(a.z, sd, p.z) - m);
    s += __expf(fmaf(a.w, sd, p.w) - m);
  }
  for (int i = (nv4 << 2) + lane; i < C; i += 32)
    s += __expf(fmaf(myn[i], sd, myp[i]) - m);
  s = wave_sum32(s);

  if (lane == 0) {
    float lse = m + __logf(s);
    int   k   = tru[b];
    float tgt = fmaf(myn[k], sd, myp[k]);
    s_red[wave] = lse - tgt;
  }
  __syncthreads();
  if (tid == 0) {
    float acc = 0.0f;
    #pragma unroll
    for (int w = 0; w < ROWS; ++w) acc += s_red[w];
    atomicAdd(&ws[2 + t], acc);
  }
}

// One wave per row b: undistorted CE term + variance depressor term.
__global__ __launch_bounds__(256) void k_undistorted(
    const float* __restrict__ pred_var, const int* __restrict__ tru,
    float* __restrict__ ws, int B, int C) {
  __shared__ __align__(16) float s_pred[ROWS * CMAX];
  __shared__ float s_red[2 * ROWS];

  const int tid  = threadIdx.x;
  const int lane = tid & 31;
  const int wave = tid >> 5;
  const int b    = blockIdx.x * ROWS + wave;
  const int C1   = C + 1;

  float* myp = &s_pred[wave * C];
  unsigned long long poff = (unsigned long long)b * (unsigned)C1 * 4ull;
  tdm_load_row(lds_offset_of(myp), (unsigned long long)(uintptr_t)pred_var + poff, C);

  const float var = pred_var[(size_t)b * C1 + C];

  __builtin_amdgcn_s_wait_tensorcnt(0);
  asm volatile("" ::: "memory");

  const float4* vp = (const float4*)myp;
  const int nv4 = C >> 2;

  float m = -INFINITY;
  for (int i = lane; i < nv4; i += 32) {
    float4 p = vp[i];
    m = fmaxf(fmaxf(m, fmaxf(p.x, p.y)), fmaxf(p.z, p.w));
  }
  for (int i = (nv4 << 2) + lane; i < C; i += 32) m = fmaxf(m, myp[i]);
  m = wave_max32(m);

  float s = 0.0f;
  for (int i = lane; i < nv4; i += 32) {
    float4 p = vp[i];
    s += __expf(p.x - m);
    s += __expf(p.y - m);
    s += __expf(p.z - m);
    s += __expf(p.w - m);
  }
  for (int i = (nv4 << 2) + lane; i < C; i += 32) s += __expf(myp[i] - m);
  s = wave_sum32(s);

  if (lane == 0) {
    float lse = m + __logf(s);
    s_red[wave]        = lse - myp[tru[b]];   // (pred+1e-15) offset cancels in lse-tgt
    s_red[ROWS + wave] = __expf(var) - 1.0f;  // variance depressor term
  }
  __syncthreads();
  if (tid == 0) {
    float a0 = 0.0f, a1 = 0.0f;
    #pragma unroll
    for (int w = 0; w < ROWS; ++w) { a0 += s_red[w]; a1 += s_red[ROWS + w]; }
    atomicAdd(&ws[0], a0);
    atomicAdd(&ws[1], a1);
  }
}

// Single-wave epilogue.
__global__ void k_final(const float* __restrict__ ws, float* __restrict__ out,
                        int B, int T) {
  const int lane = threadIdx.x;
  const float invB = 1.0f / (float)B;
  const float U    = ws[0] * invB;            // undistorted loss
  const float dep  = ws[1] * invB;            // mean variance depressor
  float mc = 0.0f;
  if (lane < T) {
    float dl = ws[2 + lane] * invB;           // distorted_loss[t]
    float x  = U - dl;
    float el = (x > 0.0f) ? x : (__expf(x) - 1.0f);   // elu(x), alpha=1
    mc = -el;                                 // monte_carlo[t]
  }
  float msum = wave_sum32(mc);
  if (lane == 0) {
    float variance_loss = (msum / (float)T) * U;
    out[0] = variance_loss + U + dep;
  }
}

extern "C" void kernel_launch(void* const* d_in, const int* in_sizes, int n_in,
                              void* d_out, int out_size, void* d_ws, size_t ws_size,
                              hipStream_t stream) {
  const float* pred_var = (const float*)d_in[0];
  const float* noise    = (const float*)d_in[1];
  const int*   tru      = (const int*)d_in[2];
  float*       out      = (float*)d_out;
  float*       ws       = (float*)d_ws;

  const int B  = in_sizes[2];
  const int C1 = in_sizes[0] / B;
  const int C  = C1 - 1;
  const int T  = (int)((long long)in_sizes[1] / ((long long)B * (long long)C));

  k_init<<<1, 64, 0, stream>>>(ws, 2 + T);
  k_undistorted<<<B / ROWS, 256, 0, stream>>>(pred_var, tru, ws, B, C);
  k_distorted<<<(B / ROWS) * T, 256, 0, stream>>>(pred_var, noise, tru, ws, B, C, T);
  k_final<<<1, 32, 0, stream>>>(ws, out, B, T);
}